// P0GCN_80942953660917
// MI455X (gfx1250) — compile-verified
//
#include <hip/hip_runtime.h>

typedef __attribute__((ext_vector_type(2))) float v2f;
typedef __attribute__((ext_vector_type(8))) float v8f;

#define N_NODES 50000
#define N_EDGES 800000
#define F_IN    96
#define F_HID   64
#define F_OUT   5
#define KEEP_P  0.4f     // 1 - DROP_P

// ---------------------------------------------------------------------------
// JAX threefry2x32 with key = jax.random.key(1) -> (k0,k1) = (0,1)
// ---------------------------------------------------------------------------
__device__ __forceinline__ unsigned rotl32(unsigned x, int d) {
    return (x << d) | (x >> (32 - d));
}

__device__ __forceinline__ void threefry2x32_01(unsigned c0, unsigned c1,
                                                unsigned& o0, unsigned& o1) {
    const unsigned k0 = 0u, k1 = 1u;
    const unsigned k2 = k0 ^ k1 ^ 0x1BD11BDAu;
    unsigned x0 = c0 + k0;
    unsigned x1 = c1 + k1;
    const int rA[4] = {13, 15, 26, 6};
    const int rB[4] = {17, 29, 16, 24};
#pragma unroll
    for (int i = 0; i < 4; ++i) { x0 += x1; x1 = rotl32(x1, rA[i]); x1 ^= x0; }
    x0 += k1; x1 += k2 + 1u;
#pragma unroll
    for (int i = 0; i < 4; ++i) { x0 += x1; x1 = rotl32(x1, rB[i]); x1 ^= x0; }
    x0 += k2; x1 += k0 + 2u;
#pragma unroll
    for (int i = 0; i < 4; ++i) { x0 += x1; x1 = rotl32(x1, rA[i]); x1 ^= x0; }
    x0 += k0; x1 += k1 + 3u;
#pragma unroll
    for (int i = 0; i < 4; ++i) { x0 += x1; x1 = rotl32(x1, rB[i]); x1 ^= x0; }
    x0 += k1; x1 += k2 + 4u;
#pragma unroll
    for (int i = 0; i < 4; ++i) { x0 += x1; x1 = rotl32(x1, rA[i]); x1 ^= x0; }
    x0 += k2; x1 += k0 + 5u;
    o0 = x0; o1 = x1;
}

// keep decision for flat index i into the (50000,64) dropout mask
__device__ __forceinline__ bool keep_mask(unsigned i) {
    const unsigned H = (unsigned)(N_NODES * F_HID) / 2u;  // 1,600,000
    unsigned j = (i < H) ? i : (i - H);
    unsigned o0, o1;
    threefry2x32_01(j, j + H, o0, o1);
    unsigned bits = (i < H) ? o0 : o1;
    unsigned fb = (bits >> 9) | 0x3f800000u;
    float u = __uint_as_float(fb) - 1.0f;
    return u < KEEP_P;
}

// ---------------------------------------------------------------------------
// Zero fill
// ---------------------------------------------------------------------------
__global__ void gcn_zero_f32(float* __restrict__ p, int n) {
    int i = blockIdx.x * blockDim.x + threadIdx.x;
    if (i < n) p[i] = 0.0f;
}

// ---------------------------------------------------------------------------
// Edge scatter-add: one wave32 per edge, F/32 floats per lane.
// Non-returning f32 atomics -> resolve in L2 (entire agg fits in 192MB L2).
// ---------------------------------------------------------------------------
template <int F>
__global__ void gcn_scatter_add(const float* __restrict__ x,
                                const int* __restrict__ src,
                                const int* __restrict__ dst,
                                float* __restrict__ agg) {
    int e = blockIdx.x * (blockDim.x >> 5) + (threadIdx.x >> 5);
    int lane = threadIdx.x & 31;
    if (e >= N_EDGES) return;               // wave-uniform exit
    int s = src[e];
    int d = dst[e];
    const float* xs = x + (size_t)s * F;
    float* ad = agg + (size_t)d * F;
#pragma unroll
    for (int j = 0; j < F / 32; ++j) {
        unsafeAtomicAdd(&ad[lane + 32 * j], xs[lane + 32 * j]);
    }
}

// ---------------------------------------------------------------------------
// GEMM1: h = dropout(relu(agg1 @ W1 + b1)).  M=50000, K=96, N=64.
// One block per 16-row M tile; 4 waves, wave w owns N columns [16w,16w+16).
// Uses V_WMMA_F32_16X16X4_F32 (exact f32), W1 staged in LDS (24KB).
// ---------------------------------------------------------------------------
__global__ __launch_bounds__(128) void gcn_gemm1_relu_drop(
    const float* __restrict__ agg1, const float* __restrict__ W1,
    const float* __restrict__ b1, float* __restrict__ h) {
    __shared__ float sW[F_IN * F_HID];  // 96*64*4B = 24KB (WGP has 320KB)
    for (int i = threadIdx.x; i < F_IN * F_HID; i += blockDim.x) sW[i] = W1[i];
    __syncthreads();

    const int wave = threadIdx.x >> 5;
    const int lane = threadIdx.x & 31;
    const int m0 = blockIdx.x * 16;
    const int n0 = wave * 16;
    const int idx = lane & 15;             // A: row m, B/C/D: col n
    const int khalf = (lane >> 4) * 2;     // K sub-offset per half-wave

    const float* arow = agg1 + (size_t)(m0 + idx) * F_IN;
    const float* brow = &sW[n0 + idx];
    v8f c = {};
#pragma unroll
    for (int k = 0; k < F_IN; k += 4) {
        v2f a, b;
        a.x = arow[k + khalf];
        a.y = arow[k + khalf + 1];
        b.x = brow[(k + khalf) * F_HID];
        b.y = brow[(k + khalf + 1) * F_HID];
        c = __builtin_amdgcn_wmma_f32_16x16x4_f32(false, a, false, b,
                                                  (short)0, c, false, false);
    }

    // epilogue: bias + relu + exact-JAX dropout mask
    const int n = n0 + idx;
    const float bias = b1[n];
    const int mbase = m0 + (lane >> 4) * 8;   // C VGPR i -> row i + 8*(lane>>4)
#pragma unroll
    for (int i = 0; i < 8; ++i) {
        int m = mbase + i;
        float v = c[i] + bias;
        v = v > 0.0f ? v : 0.0f;
        unsigned flat = (unsigned)m * (unsigned)F_HID + (unsigned)n;
        v = keep_mask(flat) ? v * (1.0f / KEEP_P) : 0.0f;
        h[(size_t)m * F_HID + n] = v;
    }
}

// ---------------------------------------------------------------------------
// GEMM2: out = agg2 @ W2 + b2.  M=50000, K=64, N=5 (padded to 16 in LDS).
// 4 M tiles per block (one per wave).
// ---------------------------------------------------------------------------
__global__ __launch_bounds__(128) void gcn_gemm2(
    const float* __restrict__ agg2, const float* __restrict__ W2,
    const float* __restrict__ b2, float* __restrict__ out) {
    __shared__ float sW[F_HID * 16];  // 64 x 16, cols 5..15 zero
    for (int i = threadIdx.x; i < F_HID * 16; i += blockDim.x) {
        int k = i >> 4, n = i & 15;
        sW[i] = (n < F_OUT) ? W2[k * F_OUT + n] : 0.0f;
    }
    __syncthreads();

    const int wave = threadIdx.x >> 5;
    const int lane = threadIdx.x & 31;
    const int mt = blockIdx.x * 4 + wave;
    if (mt >= N_NODES / 16) return;        // wave-uniform, after barrier
    const int m0 = mt * 16;
    const int idx = lane & 15;
    const int khalf = (lane >> 4) * 2;

    const float* arow = agg2 + (size_t)(m0 + idx) * F_HID;
    v8f c = {};
#pragma unroll
    for (int k = 0; k < F_HID; k += 4) {
        v2f a, b;
        a.x = arow[k + khalf];
        a.y = arow[k + khalf + 1];
        b.x = sW[(k + khalf) * 16 + idx];
        b.y = sW[(k + khalf + 1) * 16 + idx];
        c = __builtin_amdgcn_wmma_f32_16x16x4_f32(false, a, false, b,
                                                  (short)0, c, false, false);
    }

    const int n = idx;
    const int mbase = m0 + (lane >> 4) * 8;
    if (n < F_OUT) {                        // store-side masking only
        float bias = b2[n];
#pragma unroll
        for (int i = 0; i < 8; ++i) {
            out[(size_t)(mbase + i) * F_OUT + n] = c[i] + bias;
        }
    }
}

// ---------------------------------------------------------------------------
extern "C" void kernel_launch(void* const* d_in, const int* in_sizes, int n_in,
                              void* d_out, int out_size, void* d_ws, size_t ws_size,
                              hipStream_t stream) {
    const float* x  = (const float*)d_in[0];
    const int*   ei = (const int*)d_in[1];
    const float* W1 = (const float*)d_in[2];
    const float* b1 = (const float*)d_in[3];
    const float* W2 = (const float*)d_in[4];
    const float* b2 = (const float*)d_in[5];
    float* out = (float*)d_out;

    float* agg1 = (float*)d_ws;                              // 50000*96
    float* h    = agg1 + (size_t)N_NODES * F_IN;             // 50000*64
    float* agg2 = h    + (size_t)N_NODES * F_HID;            // 50000*64

    const int* src = ei;
    const int* dst = ei + N_EDGES;

    // layer 1: zero, scatter, fused GEMM+relu+dropout
    {
        int n1 = N_NODES * F_IN;
        gcn_zero_f32<<<(n1 + 255) / 256, 256, 0, stream>>>(agg1, n1);
    }
    {
        int blocks = (N_EDGES + 7) / 8;  // 8 waves (edges) per 256-thread block
        gcn_scatter_add<F_IN><<<blocks, 256, 0, stream>>>(x, src, dst, agg1);
    }
    gcn_gemm1_relu_drop<<<N_NODES / 16, 128, 0, stream>>>(agg1, W1, b1, h);

    // layer 2: zero, scatter, GEMM
    {
        int n2 = N_NODES * F_HID;
        gcn_zero_f32<<<(n2 + 255) / 256, 256, 0, stream>>>(agg2, n2);
    }
    {
        int blocks = (N_EDGES + 7) / 8;
        gcn_scatter_add<F_HID><<<blocks, 256, 0, stream>>>(h, src, dst, agg2);
    }
    gcn_gemm2<<<(N_NODES / 16 + 3) / 4, 128, 0, stream>>>(agg2, W2, b2, out);
}